// SageCox_6425271074972
// MI455X (gfx1250) — compile-verified
//
#include <hip/hip_runtime.h>

// ---------------------------------------------------------------------------
// SageCox (4x PyG SAGEConv, mean aggregation) for MI455X / gfx1250.
//
// Roofline: edge scatter is the bottleneck (~0.7 GB of L2-resident RMW,
// 192MB L2 holds the full working set); GEMMs total ~7 GFLOP -> fp32 WMMA
// (V_WMMA_F32_16X16X4_F32) at full reference precision.
//
// v2: branch-free WMMA inner loop. A and B are pre-packed into zero-padded
// buffers (Apack row-major [N][KPAD], Bpack column-major [doutpad][KPAD]) so
// the hot loop is 3x global_load_b64 + 2x v_wmma with no EXEC manipulation.
// Each wave32 owns a 32x16 output tile (two accumulators share one B fetch).
// ---------------------------------------------------------------------------

typedef float v2f __attribute__((ext_vector_type(2)));
typedef float v8f __attribute__((ext_vector_type(8)));

// ---------------- utility kernels ----------------

__global__ void zero_f32(float* __restrict__ p, long long n) {
  long long stride = (long long)gridDim.x * blockDim.x;
  for (long long i = (long long)blockIdx.x * blockDim.x + threadIdx.x; i < n;
       i += stride)
    p[i] = 0.0f;
}

__global__ void degree_kernel(const long long* __restrict__ dst,
                              float* __restrict__ deg, int E) {
  int e = blockIdx.x * blockDim.x + threadIdx.x;
  if (e < E) atomicAdd(&deg[(int)dst[e]], 1.0f);
}

// deg -> 1 / max(deg, 1)   (in place)
__global__ void rdeg_kernel(float* __restrict__ deg, int n) {
  int i = blockIdx.x * blockDim.x + threadIdx.x;
  if (i < n) deg[i] = 1.0f / fmaxf(deg[i], 1.0f);
}

// Apack[dst[e], c] += x[src[e], c]  for c in [0, DIN).
// DIN/KPAD compile-time: div/mod lower to multiply-shift. Consecutive
// threads hit consecutive channels -> coalesced on both rows.
template <int DIN, int KPAD>
__global__ void scatter_add_kernel(const float* __restrict__ x,
                                   const long long* __restrict__ src,
                                   const long long* __restrict__ dst,
                                   float* __restrict__ Apack, unsigned E) {
  const unsigned total  = E * (unsigned)DIN;
  const unsigned stride = gridDim.x * blockDim.x;
  for (unsigned i = blockIdx.x * blockDim.x + threadIdx.x; i < total;
       i += stride) {
    const unsigned e = i / (unsigned)DIN;
    const unsigned c = i - e * (unsigned)DIN;
    const int s = (int)src[e];
    const int d = (int)dst[e];
    atomicAdd(&Apack[(size_t)d * KPAD + c], x[(size_t)s * DIN + c]);
  }
}

// After scatter: cols [0,DIN) *= rdeg[n] (mean), cols [DIN,2*DIN) = x[n],
// pad cols stay zero (Apack was pre-zeroed).
template <int DIN, int KPAD>
__global__ void finalize_a_kernel(float* __restrict__ Apack,
                                  const float* __restrict__ rdeg,
                                  const float* __restrict__ x, int N) {
  const long long total  = (long long)N * KPAD;
  const long long stride = (long long)gridDim.x * blockDim.x;
  for (long long i = (long long)blockIdx.x * blockDim.x + threadIdx.x;
       i < total; i += stride) {
    const int n = (int)(i / KPAD);
    const int c = (int)(i - (long long)n * KPAD);
    if (c < DIN) {
      Apack[i] *= rdeg[n];
    } else if (c < 2 * DIN) {
      Apack[i] = x[(size_t)n * DIN + (c - DIN)];
    }
  }
}

// Bpack (column-major, zero-padded): Bpack[ncol*kpad + k] =
//   k <  din : Wl[k, ncol]
//   k < 2din : Wr[k-din, ncol]
//   else 0            (also 0 for ncol >= dout)
__global__ void pack_b_kernel(const float* __restrict__ Wl,
                              const float* __restrict__ Wr,
                              float* __restrict__ Bpack, int din, int dout,
                              int doutpad, int kpad) {
  const int total = doutpad * kpad;
  const int i = blockIdx.x * blockDim.x + threadIdx.x;
  if (i >= total) return;
  const int ncol = i / kpad;
  const int k    = i - ncol * kpad;
  float v = 0.0f;
  if (ncol < dout) {
    if (k < din)
      v = Wl[(size_t)k * dout + ncol];
    else if (k < 2 * din)
      v = Wr[(size_t)(k - din) * dout + ncol];
  }
  Bpack[i] = v;
}

// ---------------- WMMA GEMM: out = Apack @ Bpack^T(packed) + bias --------
//
// One wave32 per 32x16 output tile; K consumed 4/instr via
// V_WMMA_F32_16X16X4_F32. Inner loop is branch-free: 3 global_load_b64 +
// 2 wmma. EXEC all-ones at every WMMA (tile exit is wave-uniform).
//
// VGPR layouts (ISA 7.12.2, 32-bit):
//   A 16x4 : lanes 0-15 -> M=lane, {v0,v1}={K0,K1}; lanes 16-31 -> {K2,K3}
//   B 4x16 : lanes 0-15 -> N=lane, {v0,v1}={K0,K1}; lanes 16-31 -> {K2,K3}
//   D 16x16: VGPR r, lanes 0-15 -> M=r, lanes 16-31 -> M=8+r; N=lane&15
template <int KPAD>
__global__ void sage_gemm_wmma(const float* __restrict__ Apack,  // [N, KPAD]
                               const float* __restrict__ Bpack,  // [doutpad, KPAD]
                               const float* __restrict__ bias,   // [dout]
                               float* __restrict__ out,          // [N, dout]
                               int n_nodes, int dout, int tiles_n) {
  const int lane = threadIdx.x & 31;
  const int wave = threadIdx.x >> 5;
  const int tile = blockIdx.x * (blockDim.x >> 5) + wave;
  const int tiles_m = n_nodes >> 5;  // 32 rows per tile; N % 32 == 0
  if (tile >= tiles_m * tiles_n) return;  // wave-uniform exit
  const int tm = tile / tiles_n;
  const int tn = tile - tm * tiles_n;

  const int l15 = lane & 15;
  const int kb  = (lane >> 4) << 1;  // K sub-offset: 0 or 2

  const float* a0 = Apack + (size_t)(tm * 32 + l15) * KPAD + kb;
  const float* a1 = a0 + (size_t)16 * KPAD;
  const float* bp = Bpack + (size_t)(tn * 16 + l15) * KPAD + kb;

  v8f acc0 = {};
  v8f acc1 = {};
#pragma unroll 4
  for (int k0 = 0; k0 < KPAD; k0 += 4) {
    const v2f av0 = *(const v2f*)(a0 + k0);
    const v2f av1 = *(const v2f*)(a1 + k0);
    const v2f bv  = *(const v2f*)(bp + k0);
    // (neg_a, A, neg_b, B, c_mod, C, reuse_a, reuse_b)
    acc0 = __builtin_amdgcn_wmma_f32_16x16x4_f32(false, av0, false, bv,
                                                 (short)0, acc0, false, false);
    acc1 = __builtin_amdgcn_wmma_f32_16x16x4_f32(false, av1, false, bv,
                                                 (short)0, acc1, false, false);
  }

  const int ncol = tn * 16 + l15;
  if (ncol < dout) {
    const float bb    = bias[ncol];
    const int   mbase = tm * 32 + ((lane >> 4) << 3);  // D: M = r + 8*(lane/16)
#pragma unroll
    for (int r = 0; r < 8; ++r) {
      out[(size_t)(mbase + r) * dout + ncol]      = acc0[r] + bb;
      out[(size_t)(mbase + 16 + r) * dout + ncol] = acc1[r] + bb;
    }
  }
}

// ---------------- host orchestration ----------------

template <int DIN, int KPAD>
static void run_layer(const float* cur, const float* rdeg,
                      const long long* src, const long long* dst,
                      const float* Wl, const float* bl, const float* Wr,
                      float* Apack, float* Bpack, float* out, int N, int E,
                      int dout, int doutpad, hipStream_t stream) {
  const int T = 256;  // 8 wave32s / block

  // 1) Apack = 0
  const long long an = (long long)N * KPAD;
  zero_f32<<<(unsigned)((an + T - 1) / T), T, 0, stream>>>(Apack, an);

  // 2) scatter: Apack[dst, 0:DIN] += cur[src]
  const unsigned total = (unsigned)E * (unsigned)DIN;
  unsigned sblocks = (total + T - 1) / T;
  if (sblocks > 262144u) sblocks = 262144u;  // grid-stride covers the rest
  scatter_add_kernel<DIN, KPAD><<<sblocks, T, 0, stream>>>(cur, src, dst,
                                                           Apack, (unsigned)E);

  // 3) finalize: mean-scale + concat root features
  finalize_a_kernel<DIN, KPAD><<<(unsigned)((an + T - 1) / T), T, 0, stream>>>(
      Apack, rdeg, cur, N);

  // 4) Bpack (tiny)
  const int bn = doutpad * KPAD;
  pack_b_kernel<<<(bn + T - 1) / T, T, 0, stream>>>(Wl, Wr, Bpack, DIN, dout,
                                                    doutpad, KPAD);

  // 5) WMMA GEMM + bias
  const int tiles_n = doutpad >> 4;
  const int tiles   = (N >> 5) * tiles_n;
  const int gblocks = (tiles + (T >> 5) - 1) / (T >> 5);
  sage_gemm_wmma<KPAD><<<gblocks, T, 0, stream>>>(Apack, Bpack, bl, out, N,
                                                  dout, tiles_n);
}

extern "C" void kernel_launch(void* const* d_in, const int* in_sizes, int n_in,
                              void* d_out, int out_size, void* d_ws,
                              size_t ws_size, hipStream_t stream) {
  const float*     x0 = (const float*)d_in[0];
  const long long* ei = (const long long*)d_in[1];  // int64 [2, E]
  const int N = in_sizes[0] / 128;
  const int E = in_sizes[1] / 2;
  const long long* src = ei;
  const long long* dst = ei + E;

  const float* Wl[4] = {(const float*)d_in[2], (const float*)d_in[5],
                        (const float*)d_in[8], (const float*)d_in[11]};
  const float* bl[4] = {(const float*)d_in[3], (const float*)d_in[6],
                        (const float*)d_in[9], (const float*)d_in[12]};
  const float* Wr[4] = {(const float*)d_in[4], (const float*)d_in[7],
                        (const float*)d_in[10], (const float*)d_in[13]};

  // Workspace carve-out (256B aligned): rdeg | Apack | Bpack | bufA | bufB
  char*  ws  = (char*)d_ws;
  size_t off = 0;
  auto carve = [&](size_t bytes) -> float* {
    float* p = (float*)(ws + off);
    off = (off + bytes + 255) & ~(size_t)255;
    return p;
  };
  float* rdeg  = carve((size_t)N * sizeof(float));
  float* Apack = carve((size_t)N * 256 * sizeof(float));   // max KPAD = 256
  float* Bpack = carve((size_t)96 * 256 * sizeof(float));  // max doutpad*KPAD
  float* bufA  = carve((size_t)N * 85 * sizeof(float));
  float* bufB  = carve((size_t)N * 56 * sizeof(float));

  const int T = 256;

  // Degree + reciprocal: identical for all layers, compute once.
  zero_f32<<<(N + T - 1) / T, T, 0, stream>>>(rdeg, (long long)N);
  degree_kernel<<<(E + T - 1) / T, T, 0, stream>>>(dst, rdeg, E);
  rdeg_kernel<<<(N + T - 1) / T, T, 0, stream>>>(rdeg, N);

  // Layer pyramid: 128 -> 85 -> 56 -> 28 -> 1
  run_layer<128, 256>(x0,   rdeg, src, dst, Wl[0], bl[0], Wr[0], Apack, Bpack,
                      bufA, N, E, 85, 96, stream);
  run_layer<85, 172>(bufA,  rdeg, src, dst, Wl[1], bl[1], Wr[1], Apack, Bpack,
                     bufB, N, E, 56, 64, stream);
  run_layer<56, 112>(bufB,  rdeg, src, dst, Wl[2], bl[2], Wr[2], Apack, Bpack,
                     bufA, N, E, 28, 32, stream);
  run_layer<28, 56>(bufA,   rdeg, src, dst, Wl[3], bl[3], Wr[3], Apack, Bpack,
                    (float*)d_out, N, E, 1, 16, stream);
}